// GATSurvival_16466904613299
// MI455X (gfx1250) — compile-verified
//
#include <hip/hip_runtime.h>
#include <hip/hip_bf16.h>
#include <math.h>

typedef __attribute__((ext_vector_type(2))) float v2f;
typedef __attribute__((ext_vector_type(8))) float v8f;

#define NEG_SLOPE 0.2f
// order-preserving encoding of -inf for float atomicMax-as-uint
#define NEG_ENC 0x007FFFFFu

__device__ __forceinline__ unsigned f2ord(float f) {
  unsigned u = __float_as_uint(f);
  return (u & 0x80000000u) ? ~u : (u | 0x80000000u);
}
__device__ __forceinline__ float ord2f(unsigned u) {
  return (u & 0x80000000u) ? __uint_as_float(u & 0x7FFFFFFFu) : __uint_as_float(~u);
}

// ---------------------------------------------------------------- fills
__global__ void fill_f(float* p, float v, long long n) {
  long long t = (long long)blockIdx.x * blockDim.x + threadIdx.x;
  if (t < n) p[t] = v;
}
__global__ void fill_u(unsigned* p, unsigned v, long long n) {
  long long t = (long long)blockIdx.x * blockDim.x + threadIdx.x;
  if (t < n) p[t] = v;
}

// ---------------------------------------------------------------- fp32 WMMA GEMM
// C[M,Ncol] = A[M,K] @ B[K,Ncol], row-major. One wave computes a 16 x (16*CT)
// strip via V_WMMA_F32_16X16X4_F32. Inner loop is split load-phase /
// compute-phase so the CT B-fragments live in distinct registers: one clause
// of loads, one wait, then CT back-to-back WMMAs.
template <int CT>
__global__ void wmma_gemm_f32(const float* __restrict__ A, const float* __restrict__ B,
                              float* __restrict__ C, int M, int K, int Ncol) {
  const int wave = threadIdx.x >> 5;
  const int lane = threadIdx.x & 31;
  const int groupsN = Ncol / (16 * CT);
  const int numTiles = (M >> 4) * groupsN;
  const int tile = blockIdx.x * (blockDim.x >> 5) + wave;   // wave-uniform
  if (tile >= numTiles) return;                              // uniform exit, EXEC all-1s
  const int tm = tile / groupsN, tg = tile % groupsN;
  const int row  = tm * 16 + (lane & 15);
  const int col0 = tg * 16 * CT + (lane & 15);
  const int khi  = (lane >> 4) * 2;  // lanes 0-15 -> K{0,1}; lanes 16-31 -> K{2,3}
  v8f acc[CT];
#pragma unroll
  for (int j = 0; j < CT; ++j) acc[j] = (v8f){0.f, 0.f, 0.f, 0.f, 0.f, 0.f, 0.f, 0.f};
  const float* Arow = A + (long long)row * K;
#pragma unroll 2
  for (int k = 0; k < K; k += 4) {
    // ---- load phase: A fragment (one b64) + CT B fragments in distinct regs
    v2f a;
    a.x = Arow[k + khi];
    a.y = Arow[k + khi + 1];
    const float* B0 = B + (long long)(k + khi) * Ncol + col0;
    const float* B1 = B0 + Ncol;
    v2f bf[CT];
#pragma unroll
    for (int j = 0; j < CT; ++j) {
      bf[j].x = B0[16 * j];
      bf[j].y = B1[16 * j];
    }
    // ---- compute phase: CT consecutive WMMAs
#pragma unroll
    for (int j = 0; j < CT; ++j)
      acc[j] = __builtin_amdgcn_wmma_f32_16x16x4_f32(false, a, false, bf[j], (short)0,
                                                     acc[j], false, false);
  }
  const int mrow = tm * 16 + (lane >> 4) * 8;  // C layout: VGPR r -> M = r (+8 hi lanes)
#pragma unroll
  for (int j = 0; j < CT; ++j)
#pragma unroll
    for (int r = 0; r < 8; ++r)
      C[(long long)(mrow + r) * Ncol + col0 + 16 * j] = acc[j][r];
}

// ---------------------------------------------------------------- attention dots
__global__ void att_dots(const float* __restrict__ h, const float* __restrict__ att_s,
                         const float* __restrict__ att_d, float* __restrict__ as_,
                         float* __restrict__ ad_, int N, int H) {
  int t = blockIdx.x * blockDim.x + threadIdx.x;
  if (t >= N * H) return;
  int hh = t % H;
  const float* hv = h + (long long)t * 32;
  float s = 0.f, d = 0.f;
#pragma unroll
  for (int i = 0; i < 32; ++i) {
    float v = hv[i];
    s += v * att_s[hh * 32 + i];
    d += v * att_d[hh * 32 + i];
  }
  as_[t] = s;
  ad_[t] = d;
}

// ---------------------------------------------------------------- stage 1: logits + segment max
__global__ void edge_logit(const int* __restrict__ srcp, const int* __restrict__ dstp,
                           const float* __restrict__ as_, const float* __restrict__ ad_,
                           float* __restrict__ logit, unsigned* __restrict__ m,
                           int E, int N, int H) {
  long long t = (long long)blockIdx.x * blockDim.x + threadIdx.x;
  long long tot = (long long)(E + N) * H;
  if (t >= tot) return;
  int e = (int)(t / H), hh = (int)(t % H);
  int s = e < E ? srcp[e] : e - E;
  int d = e < E ? dstp[e] : e - E;
  float l = as_[s * H + hh] + ad_[d * H + hh];
  l = l > 0.f ? l : NEG_SLOPE * l;
  logit[t] = l;
  atomicMax(&m[d * H + hh], f2ord(l));
}

// ---------------------------------------------------------------- stage 2: exp + denominator
__global__ void edge_expsum(const int* __restrict__ dstp, float* __restrict__ alpha,
                            const unsigned* __restrict__ m, float* __restrict__ den,
                            int E, int N, int H) {
  long long t = (long long)blockIdx.x * blockDim.x + threadIdx.x;
  long long tot = (long long)(E + N) * H;
  if (t >= tot) return;
  int e = (int)(t / H), hh = (int)(t % H);
  int d = e < E ? dstp[e] : e - E;
  float a = expf(alpha[t] - ord2f(m[d * H + hh]));
  alpha[t] = a;
  atomicAdd(&den[d * H + hh], a);
}

// ---------------------------------------------------------------- stage 3: normalize
__global__ void edge_norm(const int* __restrict__ dstp, float* __restrict__ alpha,
                          const float* __restrict__ den, int E, int N, int H) {
  long long t = (long long)blockIdx.x * blockDim.x + threadIdx.x;
  long long tot = (long long)(E + N) * H;
  if (t >= tot) return;
  int e = (int)(t / H), hh = (int)(t % H);
  int d = e < E ? dstp[e] : e - E;
  alpha[t] = alpha[t] / den[d * H + hh];
}

// ---------------------------------------------------------------- stage 4: gather-scale-scatter
// one thread per (edge, h, d); coef load is wave-uniform (broadcast), h gather is
// a coalesced 128B line per wave, one f32 atomic-add scatter per thread.
__global__ void edge_agg(const int* __restrict__ srcp, const int* __restrict__ dstp,
                         const float* __restrict__ alpha, const float* __restrict__ h,
                         float* __restrict__ out, int E, int N, int H) {
  long long t = (long long)blockIdx.x * blockDim.x + threadIdx.x;
  const int F = H * 32;
  long long tot = (long long)(E + N) * F;
  if (t >= tot) return;
  int f = (int)(t % F);
  int e = (int)(t / F);
  int hh = f >> 5;
  int s = e < E ? srcp[e] : e - E;
  int d = e < E ? dstp[e] : e - E;
  float coef = alpha[(long long)e * H + hh];
  atomicAdd(&out[(long long)d * F + f], coef * h[(long long)s * F + f]);
}

// ---------------------------------------------------------------- bias + ELU (in place)
__global__ void bias_elu(float* __restrict__ x, const float* __restrict__ b,
                         long long N, int F) {
  long long t = (long long)blockIdx.x * blockDim.x + threadIdx.x;
  if (t >= N * F) return;
  int f = (int)(t % F);
  float v = x[t] + b[f];
  x[t] = v > 0.f ? v : (expf(v) - 1.f);
}

// ---------------------------------------------------------------- global mean pool (sum stage)
__global__ void pool_sum(const float* __restrict__ h, const int* __restrict__ batch,
                         float* __restrict__ sums, float* __restrict__ cnt, int N) {
  int t = blockIdx.x * blockDim.x + threadIdx.x;
  if (t >= N * 32) return;
  int n = t >> 5, d = t & 31;
  int g = batch[n];
  atomicAdd(&sums[g * 32 + d], h[t]);
  if (d == 0) atomicAdd(&cnt[g], 1.f);
}

// ---------------------------------------------------------------- tiny classifier MLP
__global__ void mlp_head(const float* __restrict__ sums, const float* __restrict__ cnt,
                         const float* __restrict__ Wc1, const float* __restrict__ bc1,
                         const float* __restrict__ Wc2, const float* __restrict__ bc2,
                         float* __restrict__ out, int G) {
  int g = threadIdx.x;
  if (g >= G) return;
  float c = cnt[g];
  c = c < 1.f ? 1.f : c;
  float gv[32];
#pragma unroll
  for (int d = 0; d < 32; ++d) gv[d] = sums[g * 32 + d] / c;
  float r = bc2[0];
  for (int j = 0; j < 16; ++j) {
    float hsum = bc1[j];
#pragma unroll
    for (int d = 0; d < 32; ++d) hsum += gv[d] * Wc1[d * 16 + j];
    hsum = hsum > 0.f ? hsum : 0.f;
    r += hsum * Wc2[j];
  }
  out[g] = r;
}

// ================================================================ host
static inline unsigned nblk(long long n, int b) { return (unsigned)((n + b - 1) / b); }

extern "C" void kernel_launch(void* const* d_in, const int* in_sizes, int n_in,
                              void* d_out, int out_size, void* d_ws, size_t ws_size,
                              hipStream_t stream) {
  const float* x     = (const float*)d_in[0];
  const int*   ei    = (const int*)d_in[1];
  const int*   batch = (const int*)d_in[2];
  const float* W1    = (const float*)d_in[3];
  const float* aS1   = (const float*)d_in[4];
  const float* aD1   = (const float*)d_in[5];
  const float* b1    = (const float*)d_in[6];
  const float* W2    = (const float*)d_in[7];
  const float* aS2   = (const float*)d_in[8];
  const float* aD2   = (const float*)d_in[9];
  const float* b2    = (const float*)d_in[10];
  const float* Wc1   = (const float*)d_in[11];
  const float* bc1   = (const float*)d_in[12];
  const float* Wc2   = (const float*)d_in[13];
  const float* bc2   = (const float*)d_in[14];

  const int E = in_sizes[1] / 2;      // 800000
  const int N = in_sizes[2];          // 50000
  const int H1 = 8, F1 = 256, G = 32;
  const long long ET = (long long)E + N;   // edges incl. self loops

  const int* srcp = ei;
  const int* dstp = ei + E;

  // ---- workspace carve-out
  char* w = (char*)d_ws;
  size_t o = 0;
  auto carve = [&](size_t bytes) -> void* {
    void* p = w + o;
    o += (bytes + 255) & ~(size_t)255;
    return p;
  };
  float*    h1    = (float*)carve((size_t)N * F1 * 4);
  float*    out1  = (float*)carve((size_t)N * F1 * 4);
  float*    alpha = (float*)carve((size_t)ET * H1 * 4);   // reused by layer 2 (H=1)
  float*    as1   = (float*)carve((size_t)N * H1 * 4);
  float*    ad1   = (float*)carve((size_t)N * H1 * 4);
  unsigned* m1    = (unsigned*)carve((size_t)N * H1 * 4);
  float*    den1  = (float*)carve((size_t)N * H1 * 4);
  float*    h2    = (float*)carve((size_t)N * 32 * 4);
  float*    out2  = (float*)carve((size_t)N * 32 * 4);
  float*    as2   = (float*)carve((size_t)N * 4);
  float*    ad2   = (float*)carve((size_t)N * 4);
  unsigned* m2    = (unsigned*)carve((size_t)N * 4);
  float*    den2  = (float*)carve((size_t)N * 4);
  float*    sums  = (float*)carve((size_t)G * 32 * 4);
  float*    cnt   = (float*)carve((size_t)G * 4);

  const int T = 256;

  // ---- init accumulators
  fill_f<<<nblk((long long)N * F1, T), T, 0, stream>>>(out1, 0.f, (long long)N * F1);
  fill_u<<<nblk((long long)N * H1, T), T, 0, stream>>>(m1, NEG_ENC, (long long)N * H1);
  fill_f<<<nblk((long long)N * H1, T), T, 0, stream>>>(den1, 0.f, (long long)N * H1);
  fill_f<<<nblk((long long)N * 32, T), T, 0, stream>>>(out2, 0.f, (long long)N * 32);
  fill_u<<<nblk(N, T), T, 0, stream>>>(m2, NEG_ENC, N);
  fill_f<<<nblk(N, T), T, 0, stream>>>(den2, 0.f, N);
  fill_f<<<1, T, 0, stream>>>(sums, 0.f, G * 32);
  fill_f<<<1, G, 0, stream>>>(cnt, 0.f, G);

  // ---- layer 1: projection GEMM (fp32 WMMA), [N,256] = [N,256] @ [256,256]
  {
    int tiles = (N / 16) * (F1 / 64);          // 16x64 strip per wave (CT=4)
    wmma_gemm_f32<4><<<nblk(tiles, 4), 128, 0, stream>>>(x, W1, h1, N, F1, F1);
  }
  att_dots<<<nblk((long long)N * H1, T), T, 0, stream>>>(h1, aS1, aD1, as1, ad1, N, H1);
  edge_logit<<<nblk(ET * H1, T), T, 0, stream>>>(srcp, dstp, as1, ad1, alpha, m1, E, N, H1);
  edge_expsum<<<nblk(ET * H1, T), T, 0, stream>>>(dstp, alpha, m1, den1, E, N, H1);
  edge_norm<<<nblk(ET * H1, T), T, 0, stream>>>(dstp, alpha, den1, E, N, H1);
  edge_agg<<<nblk(ET * F1, T), T, 0, stream>>>(srcp, dstp, alpha, h1, out1, E, N, H1);
  bias_elu<<<nblk((long long)N * F1, T), T, 0, stream>>>(out1, b1, N, F1);

  // ---- layer 2: projection GEMM (fp32 WMMA), [N,32] = [N,256] @ [256,32]
  {
    int tiles = (N / 16) * (32 / 32);          // 16x32 strip per wave (CT=2)
    wmma_gemm_f32<2><<<nblk(tiles, 4), 128, 0, stream>>>(out1, W2, h2, N, F1, 32);
  }
  att_dots<<<nblk(N, T), T, 0, stream>>>(h2, aS2, aD2, as2, ad2, N, 1);
  edge_logit<<<nblk(ET, T), T, 0, stream>>>(srcp, dstp, as2, ad2, alpha, m2, E, N, 1);
  edge_expsum<<<nblk(ET, T), T, 0, stream>>>(dstp, alpha, m2, den2, E, N, 1);
  edge_norm<<<nblk(ET, T), T, 0, stream>>>(dstp, alpha, den2, E, N, 1);
  edge_agg<<<nblk(ET * 32, T), T, 0, stream>>>(srcp, dstp, alpha, h2, out2, E, N, 1);
  bias_elu<<<nblk((long long)N * 32, T), T, 0, stream>>>(out2, b2, N, 32);

  // ---- global mean pool + classifier head
  pool_sum<<<nblk((long long)N * 32, T), T, 0, stream>>>(out2, batch, sums, cnt, N);
  mlp_head<<<1, 32, 0, stream>>>(sums, cnt, Wc1, bc1, Wc2, bc2, (float*)d_out, G);
}